// LightGCN_39453569581264
// MI455X (gfx1250) — compile-verified
//
#include <hip/hip_runtime.h>
#include <hip/hip_bf16.h>

#define EMB            64
#define V4ROW          16      // 64 floats = 16 float4 per row
#define TPB            256
#define ROWS_PER_BLOCK 16      // 16 lanes per row -> 16 rows per 256-thread block
#define CHUNK          512     // edges staged in LDS per iteration (512*8B = 4KB)
#define SCAN_T         1024

// ---------------- CDNA5 async global->LDS staging (guarded) ----------------
#if defined(__HIP_DEVICE_COMPILE__) && __has_builtin(__builtin_amdgcn_global_load_async_to_lds_b32)
#define HAVE_ASYNC_LDS 1
#else
#define HAVE_ASYNC_LDS 0
#endif

typedef __attribute__((address_space(1))) int gas_int;
typedef __attribute__((address_space(3))) int las_int;

__device__ __forceinline__ void async_copy_b32(void* lds_dst, const void* g_src) {
#if HAVE_ASYNC_LDS
  __builtin_amdgcn_global_load_async_to_lds_b32(
      (gas_int*)g_src,
      (las_int*)lds_dst,
      0, 0);
#else
  *(int*)lds_dst = *(const int*)g_src;   // bitwise b32 copy fallback
#endif
}

__device__ __forceinline__ void wait_async_lds() {
#if HAVE_ASYNC_LDS
#if __has_builtin(__builtin_amdgcn_s_wait_asynccnt)
  __builtin_amdgcn_s_wait_asynccnt(0);
#else
  asm volatile("s_wait_asynccnt 0" ::: "memory");
#endif
#endif
}

// ---------------- small utility kernels ----------------
__global__ void lg_init_concat(const float4* __restrict__ ue,
                               const float4* __restrict__ ie,
                               float4* __restrict__ x0,
                               float4* __restrict__ acc,
                               int userV4, int totalV4) {
  int i = blockIdx.x * blockDim.x + threadIdx.x;
  if (i < totalV4) {
    float4 v = (i < userV4) ? ue[i] : ie[i - userV4];
    x0[i]  = v;
    acc[i] = v;
  }
}

__global__ void lg_zero_i32(int* __restrict__ p, int n) {
  int i = blockIdx.x * blockDim.x + threadIdx.x;
  if (i < n) p[i] = 0;
}

__global__ void lg_zero_f4(float4* __restrict__ p, int n) {
  int i = blockIdx.x * blockDim.x + threadIdx.x;
  if (i < n) p[i] = make_float4(0.f, 0.f, 0.f, 0.f);
}

__global__ void lg_copy_i32(const int* __restrict__ s, int* __restrict__ d, int n) {
  int i = blockIdx.x * blockDim.x + threadIdx.x;
  if (i < n) d[i] = s[i];
}

__global__ void lg_histogram(const int* __restrict__ rows, int* __restrict__ counts, int nnz) {
  int e = blockIdx.x * blockDim.x + threadIdx.x;
  if (e < nnz) atomicAdd(&counts[rows[e]], 1);
}

// single-workgroup chunked exclusive scan: ptr[0..n] from counts[0..n-1]
__global__ __launch_bounds__(SCAN_T) void lg_scan(const int* __restrict__ counts,
                                                  int* __restrict__ ptr, int n) {
  __shared__ int buf[SCAN_T];
  __shared__ int carry;
  const int tid = threadIdx.x;
  if (tid == 0) carry = 0;
  __syncthreads();
  for (int base = 0; base < n; base += SCAN_T) {
    const int idx = base + tid;
    const int v = (idx < n) ? counts[idx] : 0;
    buf[tid] = v;
    __syncthreads();
    for (int off = 1; off < SCAN_T; off <<= 1) {
      int t = (tid >= off) ? buf[tid - off] : 0;
      __syncthreads();
      buf[tid] += t;
      __syncthreads();
    }
    if (idx < n) ptr[idx] = carry + buf[tid] - v;   // exclusive
    __syncthreads();
    if (tid == SCAN_T - 1) carry += buf[SCAN_T - 1];
    __syncthreads();
  }
  if (tid == 0) ptr[n] = carry;
}

__global__ void lg_scatter(const int* __restrict__ rows,
                           const int* __restrict__ cols,
                           const float* __restrict__ vals,
                           int* __restrict__ cursor,
                           int* __restrict__ cols_s,
                           float* __restrict__ vals_s, int nnz) {
  int e = blockIdx.x * blockDim.x + threadIdx.x;
  if (e < nnz) {
    int r = rows[e];
    int p = atomicAdd(&cursor[r], 1);
    cols_s[p] = cols[e];
    vals_s[p] = vals[e];
  }
}

// ---------------- atomic-free CSR SpMM, fused with running accumulation ----------------
__global__ __launch_bounds__(TPB) void lg_spmm_csr(
    const int*    __restrict__ row_ptr,
    const int*    __restrict__ cols_s,
    const float*  __restrict__ vals_s,
    const float4* __restrict__ x,      // [N * V4ROW]
    float4*       __restrict__ x_new,  // [N * V4ROW]
    float4*       __restrict__ acc,    // d_out as float4, acc = (acc + x_new) * scale
    float scale, int n_rows)
{
  __shared__ int   s_col[CHUNK];
  __shared__ float s_val[CHUNK];

  const int row0   = blockIdx.x * ROWS_PER_BLOCK;
  const int lane16 = threadIdx.x & 15;
  const int rloc   = threadIdx.x >> 4;
  const int row    = row0 + rloc;
  if (row0 >= n_rows) return;

  const int eStart = row_ptr[row0];
  const int rowHi  = (row0 + ROWS_PER_BLOCK < n_rows) ? (row0 + ROWS_PER_BLOCK) : n_rows;
  const int eEnd   = row_ptr[rowHi];

  int myStart = eEnd, myEnd = eEnd;
  if (row < n_rows) { myStart = row_ptr[row]; myEnd = row_ptr[row + 1]; }

  float4 sum = make_float4(0.f, 0.f, 0.f, 0.f);

  for (int base = eStart; base < eEnd; base += CHUNK) {
    const int cnt = (eEnd - base < CHUNK) ? (eEnd - base) : CHUNK;
    // cooperatively stage this block's contiguous edge segment into LDS (async on CDNA5)
    for (int i = threadIdx.x; i < cnt; i += TPB) {
      async_copy_b32(&s_col[i], &cols_s[base + i]);
      async_copy_b32(&s_val[i], &vals_s[base + i]);
    }
    wait_async_lds();
    __syncthreads();

    const int lo = (myStart > base) ? myStart : base;
    const int hi = (myEnd < base + cnt) ? myEnd : (base + cnt);
    for (int e = lo; e < hi; ++e) {
      if (e + 4 < hi) {   // prefetch the gather target a few edges ahead
        const int cp = s_col[e + 4 - base];
        __builtin_prefetch(&x[(size_t)cp * V4ROW + lane16], 0, 1);
      }
      const int   c = s_col[e - base];
      const float v = s_val[e - base];
      const float4 m = x[(size_t)c * V4ROW + lane16];
      sum.x = fmaf(v, m.x, sum.x);
      sum.y = fmaf(v, m.y, sum.y);
      sum.z = fmaf(v, m.z, sum.z);
      sum.w = fmaf(v, m.w, sum.w);
    }
    __syncthreads();
  }

  if (row < n_rows) {
    const size_t o = (size_t)row * V4ROW + lane16;
    x_new[o] = sum;
    float4 a = acc[o];
    a.x = (a.x + sum.x) * scale;
    a.y = (a.y + sum.y) * scale;
    a.z = (a.z + sum.z) * scale;
    a.w = (a.w + sum.w) * scale;
    acc[o] = a;
  }
}

// ---------------- fallback COO path (only if ws too small for CSR scratch) ----------------
__global__ void lg_spmm_atomic(const int* __restrict__ rows,
                               const int* __restrict__ cols,
                               const float* __restrict__ vals,
                               const float4* __restrict__ x,
                               float* __restrict__ x_new, int nnz) {
  long long t = (long long)blockIdx.x * blockDim.x + threadIdx.x;
  int e = (int)(t >> 4);
  int lane16 = (int)(t & 15);
  if (e >= nnz) return;
  int r = rows[e], c = cols[e];
  float v = vals[e];
  float4 m = x[(size_t)c * V4ROW + lane16];
  float* dst = x_new + (size_t)r * EMB + lane16 * 4;
  atomicAdd(dst + 0, v * m.x);
  atomicAdd(dst + 1, v * m.y);
  atomicAdd(dst + 2, v * m.z);
  atomicAdd(dst + 3, v * m.w);
}

__global__ void lg_acc_combine(float4* __restrict__ acc, const float4* __restrict__ xnew,
                               float scale, int nV4) {
  int i = blockIdx.x * blockDim.x + threadIdx.x;
  if (i < nV4) {
    float4 a = acc[i], s = xnew[i];
    a.x = (a.x + s.x) * scale;
    a.y = (a.y + s.y) * scale;
    a.z = (a.z + s.z) * scale;
    a.w = (a.w + s.w) * scale;
    acc[i] = a;
  }
}

static inline int divUp(long long a, long long b) { return (int)((a + b - 1) / b); }

extern "C" void kernel_launch(void* const* d_in, const int* in_sizes, int n_in,
                              void* d_out, int out_size, void* d_ws, size_t ws_size,
                              hipStream_t stream) {
  const float* user_emb = (const float*)d_in[0];
  const float* item_emb = (const float*)d_in[1];
  const float* adj_vals = (const float*)d_in[2];
  const int*   adj_rows = (const int*)d_in[3];
  const int*   adj_cols = (const int*)d_in[4];

  const int n_users = in_sizes[0] / EMB;
  const int n_items = in_sizes[1] / EMB;
  const int N       = n_users + n_items;
  const int nnz     = in_sizes[2];
  const int totalV4 = N * V4ROW;
  const int userV4  = n_users * V4ROW;

  // workspace layout
  size_t off = 0;
  float* Xa = (float*)((char*)d_ws + off); off += (size_t)N * EMB * 4;
  float* Xb = (float*)((char*)d_ws + off); off += (size_t)N * EMB * 4;
  const size_t atomicNeed = off;
  int* row_ptr = (int*)((char*)d_ws + off); off += (size_t)(N + 1) * 4;
  int* cursor  = (int*)((char*)d_ws + off); off += (size_t)N * 4;
  off = (off + 15) & ~(size_t)15;
  int*   cols_s = (int*)((char*)d_ws + off);   off += (size_t)nnz * 4;
  float* vals_s = (float*)((char*)d_ws + off); off += (size_t)nnz * 4;
  const size_t csrNeed = off;

  float4* out4 = (float4*)d_out;

  // acc (= d_out) and X0 both initialized to concat(user_emb, item_emb)
  lg_init_concat<<<divUp(totalV4, TPB), TPB, 0, stream>>>(
      (const float4*)user_emb, (const float4*)item_emb,
      (float4*)Xa, out4, userV4, totalV4);

  if (ws_size >= csrNeed) {
    // ---- build CSR once (structure is layer-invariant) ----
    lg_zero_i32<<<divUp(N, TPB), TPB, 0, stream>>>(cursor, N);          // cursor doubles as counts
    lg_histogram<<<divUp(nnz, TPB), TPB, 0, stream>>>(adj_rows, cursor, nnz);
    lg_scan<<<1, SCAN_T, 0, stream>>>(cursor, row_ptr, N);
    lg_copy_i32<<<divUp(N, TPB), TPB, 0, stream>>>(row_ptr, cursor, N);
    lg_scatter<<<divUp(nnz, TPB), TPB, 0, stream>>>(adj_rows, adj_cols, adj_vals,
                                                    cursor, cols_s, vals_s, nnz);

    // ---- 3 atomic-free SpMM layers, fused accumulation, final /4 fused into layer 3 ----
    const int spmmBlocks = divUp(N, ROWS_PER_BLOCK);
    lg_spmm_csr<<<spmmBlocks, TPB, 0, stream>>>(row_ptr, cols_s, vals_s,
        (const float4*)Xa, (float4*)Xb, out4, 1.0f,  N);
    lg_spmm_csr<<<spmmBlocks, TPB, 0, stream>>>(row_ptr, cols_s, vals_s,
        (const float4*)Xb, (float4*)Xa, out4, 1.0f,  N);
    lg_spmm_csr<<<spmmBlocks, TPB, 0, stream>>>(row_ptr, cols_s, vals_s,
        (const float4*)Xa, (float4*)Xb, out4, 0.25f, N);
  } else {
    // ---- fallback: COO with float atomics (only needs the two X buffers) ----
    (void)atomicNeed;
    const int edgeBlocks = divUp((long long)nnz * 16, TPB);
    float* xo = Xa; float* xn = Xb;
    for (int layer = 0; layer < 3; ++layer) {
      lg_zero_f4<<<divUp(totalV4, TPB), TPB, 0, stream>>>((float4*)xn, totalV4);
      lg_spmm_atomic<<<edgeBlocks, TPB, 0, stream>>>(adj_rows, adj_cols, adj_vals,
                                                     (const float4*)xo, xn, nnz);
      lg_acc_combine<<<divUp(totalV4, TPB), TPB, 0, stream>>>(
          out4, (const float4*)xn, (layer == 2) ? 0.25f : 1.0f, totalV4);
      float* t = xo; xo = xn; xn = t;
    }
  }
}